// RegimeSwitchingKF_74448963109261
// MI455X (gfx1250) — compile-verified
//
#include <hip/hip_runtime.h>
#include <cstddef>

#define O_DIM 9
#define L_DIM 4
#define NSER 8192
#define NT_STEPS 400
#define LOG2PI_F 1.8378770664093453f

typedef __attribute__((ext_vector_type(2))) float v2f;
typedef __attribute__((ext_vector_type(8))) float v8f;

// ---------------------------------------------------------------------------
// Per-regime Kalman update. Regime R==1: L1 row o selects state o/3 with
// coefficient c[o]. Regime R==2: every row selects state 3 with coeff c[o].
// All loops fully unrolled; matrices live in VGPRs.
// ---------------------------------------------------------------------------
template <int R>
__device__ __forceinline__ constexpr int sidx_of(int o) {
    return (R == 1) ? (o / 3) : 3;
}

template <int R>
__device__ __forceinline__ void regime_update(
    const float Pp[L_DIM][L_DIM], const float ep[L_DIM],
    const float v[O_DIM], const float c[O_DIM], const float Rd[O_DIM],
    float* lik_out, float eta_u[L_DIM], float Pu[L_DIM][L_DIM])
{
    // F = L * Pp * L^T + diag(Rd) + 1e-6*I   (lower triangle only)
    float F[O_DIM][O_DIM];
#pragma unroll
    for (int o = 0; o < O_DIM; ++o) {
#pragma unroll
        for (int p = 0; p <= o; ++p) {
            float f = c[o] * c[p] * Pp[sidx_of<R>(o)][sidx_of<R>(p)];
            if (o == p) f += Rd[o] + 1e-6f;
            F[o][p] = f;
        }
    }
    // In-place Cholesky (lower), keep reciprocal diagonal.
    float invd[O_DIM];
#pragma unroll
    for (int j = 0; j < O_DIM; ++j) {
        float d = F[j][j];
#pragma unroll
        for (int k = 0; k < j; ++k) d -= F[j][k] * F[j][k];
        float g = sqrtf(d);
        F[j][j] = g;
        float inv = 1.0f / g;
        invd[j] = inv;
#pragma unroll
        for (int i = j + 1; i < O_DIM; ++i) {
            float s = F[i][j];
#pragma unroll
            for (int k = 0; k < j; ++k) s -= F[i][k] * F[j][k];
            F[i][j] = s * inv;
        }
    }
    // z = G^{-1} v ; Gaussian log-likelihood of v under N(0, F)
    float z[O_DIM];
    float zz = 0.0f, hld = 0.0f;
#pragma unroll
    for (int o = 0; o < O_DIM; ++o) {
        float s = v[o];
#pragma unroll
        for (int k = 0; k < o; ++k) s -= F[o][k] * z[k];
        z[o] = s * invd[o];
        zz += z[o] * z[o];
        hld += logf(F[o][o]);
    }
    float ll = -0.5f * (zz + (float)O_DIM * LOG2PI_F) - hld;
    *lik_out = expf(ll);

    // K = (F^{-1} (L Pp))^T, solved column-by-column via the Cholesky factor.
    float K[L_DIM][O_DIM];
#pragma unroll
    for (int l = 0; l < L_DIM; ++l) {
        float w[O_DIM];
#pragma unroll
        for (int o = 0; o < O_DIM; ++o) {
            float s = c[o] * Pp[sidx_of<R>(o)][l];
#pragma unroll
            for (int k = 0; k < o; ++k) s -= F[o][k] * w[k];
            w[o] = s * invd[o];
        }
#pragma unroll
        for (int o = O_DIM - 1; o >= 0; --o) {
            float s = w[o];
#pragma unroll
            for (int k = o + 1; k < O_DIM; ++k) s -= F[k][o] * w[k];
            w[o] = s * invd[o];
            K[l][o] = w[o];
        }
    }
    // eta_u = ep + K v
#pragma unroll
    for (int l = 0; l < L_DIM; ++l) {
        float s = ep[l];
#pragma unroll
        for (int o = 0; o < O_DIM; ++o) s += K[l][o] * v[o];
        eta_u[l] = s;
    }
    // M = I - K L  (L sparse: row o hits column sidx(o) with coeff c[o])
    float M[L_DIM][L_DIM];
#pragma unroll
    for (int l = 0; l < L_DIM; ++l) {
#pragma unroll
        for (int m = 0; m < L_DIM; ++m) {
            float s = 0.0f;
#pragma unroll
            for (int o = 0; o < O_DIM; ++o)
                if (sidx_of<R>(o) == m) s += K[l][o] * c[o];
            M[l][m] = ((l == m) ? 1.0f : 0.0f) - s;
        }
    }
    // Pu = M Pp M^T + K diag(Rd) K^T
    float T[L_DIM][L_DIM];
#pragma unroll
    for (int l = 0; l < L_DIM; ++l)
#pragma unroll
        for (int m = 0; m < L_DIM; ++m) {
            float s = 0.0f;
#pragma unroll
            for (int k = 0; k < L_DIM; ++k) s += M[l][k] * Pp[k][m];
            T[l][m] = s;
        }
#pragma unroll
    for (int l = 0; l < L_DIM; ++l)
#pragma unroll
        for (int m = 0; m < L_DIM; ++m) {
            float s = 0.0f;
#pragma unroll
            for (int k = 0; k < L_DIM; ++k) s += T[l][k] * M[m][k];
#pragma unroll
            for (int o = 0; o < O_DIM; ++o) s += K[l][o] * Rd[o] * K[m][o];
            Pu[l][m] = s;
        }
}

// ---------------------------------------------------------------------------
// Main filter: one lane = one series; serial scan over 400 timesteps.
// ---------------------------------------------------------------------------
__global__ __launch_bounds__(256, 1)
void rskf_forward_kernel(const float* __restrict__ y,
                         const float* __restrict__ B1s1,
                         const float* __restrict__ B1s2,
                         const float* __restrict__ lam1,
                         const float* __restrict__ lam2,
                         const float* __restrict__ logq,
                         const float* __restrict__ logr,
                         const float* __restrict__ gint,
                         const float* __restrict__ gcoef,
                         float* __restrict__ block_partials)
{
    const int tid = threadIdx.x;
    const int n = blockIdx.x * blockDim.x + tid;

    // ---- model parameters (uniform across lanes) ----
    float B[4][4];
#pragma unroll
    for (int i = 0; i < 4; ++i)
#pragma unroll
        for (int j = 0; j < 4; ++j) B[i][j] = 0.0f;
#pragma unroll
    for (int i = 0; i < 3; ++i)
#pragma unroll
        for (int j = 0; j < 3; ++j) B[i][j] = B1s1[i * 3 + j];
    B[3][3] = B1s2[0];

    float Qd[4], Rd[9];
#pragma unroll
    for (int i = 0; i < 4; ++i) Qd[i] = expf(logq[i]);
#pragma unroll
    for (int o = 0; o < 9; ++o) Rd[o] = expf(logr[o]);

    float c1[9];
    c1[0] = 1.0f; c1[1] = lam1[0]; c1[2] = lam1[1];
    c1[3] = 1.0f; c1[4] = lam1[2]; c1[5] = lam1[3];
    c1[6] = 1.0f; c1[7] = lam1[4]; c1[8] = lam1[5];
    float c2[9];
    c2[0] = 1.0f;
#pragma unroll
    for (int o = 1; o < 9; ++o) c2[o] = lam2[o - 1];

    const float g0 = gint[0];
    const float gc0 = gcoef[0], gc1 = gcoef[1], gc2 = gcoef[2];

    // ---- filter state ----
    float prob0 = 0.99f, prob1 = 0.01f;
    float eta[4] = {0.0f, 0.0f, 0.0f, 0.0f};
    float P[4][4];
#pragma unroll
    for (int i = 0; i < 4; ++i)
#pragma unroll
        for (int j = 0; j < 4; ++j) P[i][j] = (i == j) ? 1000.0f : 0.0f;

    float acc = 0.0f;
    const float* yrow = y + (size_t)n * (size_t)(NT_STEPS * O_DIM);

    for (int t = 0; t < NT_STEPS; ++t) {
        const float* yp = yrow + t * O_DIM;
        float yt[9];
#pragma unroll
        for (int o = 0; o < 9; ++o) yt[o] = yp[o];
        __builtin_prefetch(yp + O_DIM, 0, 1);   // next timestep -> global_prefetch

        // regime transition prob (state 1 -> 1)
        float logit = g0 + eta[0] * gc0 + eta[1] * gc1 + eta[2] * gc2;
        float p11 = 1.0f / (1.0f + expf(-logit));
        float p12 = 1.0f - p11;

        // predict: ep = B eta ; Pp = B P B^T + diag(Qd)
        float ep[4];
#pragma unroll
        for (int i = 0; i < 4; ++i) {
            float s = 0.0f;
#pragma unroll
            for (int k = 0; k < 4; ++k) s += B[i][k] * eta[k];
            ep[i] = s;
        }
        float T[4][4], Pp[4][4];
#pragma unroll
        for (int i = 0; i < 4; ++i)
#pragma unroll
            for (int j = 0; j < 4; ++j) {
                float s = 0.0f;
#pragma unroll
                for (int k = 0; k < 4; ++k) s += B[i][k] * P[k][j];
                T[i][j] = s;
            }
#pragma unroll
        for (int i = 0; i < 4; ++i)
#pragma unroll
            for (int j = 0; j < 4; ++j) {
                float s = (i == j) ? Qd[i] : 0.0f;
#pragma unroll
                for (int k = 0; k < 4; ++k) s += T[i][k] * B[j][k];
                Pp[i][j] = s;
            }

        // innovations
        float v1[9], v2[9];
#pragma unroll
        for (int o = 0; o < 9; ++o) {
            v1[o] = yt[o] - c1[o] * ep[sidx_of<1>(o)];
            v2[o] = yt[o] - c2[o] * ep[3];
        }

        float lik1, lik2;
        float e1[4], e2[4];
        float P1[4][4], P2[4][4];
        regime_update<1>(Pp, ep, v1, c1, Rd, &lik1, e1, P1);
        regime_update<2>(Pp, ep, v2, c2, Rd, &lik2, e2, P2);

        // regime mixing (p21 = 0, p22 = 1)
        float num1 = lik1 * (prob0 * p11);
        float num2 = lik2 * (prob0 * p12 + prob1);
        float marg = num1 + num2 + 1e-9f;
        float pt1 = num1 / marg;
        float pt2 = num2 / marg;
        acc += logf(marg);

        float eta_t[4];
#pragma unroll
        for (int i = 0; i < 4; ++i) eta_t[i] = pt1 * e1[i] + pt2 * e2[i];
        float d1[4], d2[4];
#pragma unroll
        for (int i = 0; i < 4; ++i) { d1[i] = e1[i] - eta_t[i]; d2[i] = e2[i] - eta_t[i]; }
#pragma unroll
        for (int i = 0; i < 4; ++i)
#pragma unroll
            for (int j = 0; j < 4; ++j)
                P[i][j] = pt1 * (P1[i][j] + d1[i] * d1[j]) +
                          pt2 * (P2[i][j] + d2[i] * d2[j]);
#pragma unroll
        for (int i = 0; i < 4; ++i) eta[i] = eta_t[i];
        prob0 = pt1;
        prob1 = pt2;
    }

    // deterministic in-block tree reduction
    __shared__ float sh[256];
    sh[tid] = acc;
    __syncthreads();
#pragma unroll
    for (int s = 128; s > 0; s >>= 1) {
        if (tid < s) sh[tid] += sh[tid + s];
        __syncthreads();
    }
    if (tid == 0) block_partials[blockIdx.x] = sh[0];
}

// ---------------------------------------------------------------------------
// Final reduction of the 32 block partials in a single wave using the full-f32
// WMMA shape V_WMMA_F32_16X16X4_F32 (fixed hardware accumulation order ->
// deterministic). A(16x4): lane m supplies A[m%16][K], K=0 for lanes 0-15,
// K=2 for lanes 16-31 (K=1,3 zero). B = ones(4x16). D[m][n] = x_m + x_{m+16}.
// Column sum of D + one cross-half shuffle = total.
// ---------------------------------------------------------------------------
__global__ void rskf_final_reduce(const float* __restrict__ parts,
                                  float* __restrict__ out, int nparts)
{
    int lane = threadIdx.x;              // 32 threads = 1 wave, EXEC all ones
    float x = (lane < nparts) ? parts[lane] : 0.0f;
    v2f a; a.x = x;    a.y = 0.0f;
    v2f b; b.x = 1.0f; b.y = 1.0f;
    v8f c = {};
    c = __builtin_amdgcn_wmma_f32_16x16x4_f32(
            /*neg_a=*/false, a, /*neg_b=*/false, b,
            /*c_mod=*/(short)0, c, /*reuse_a=*/false, /*reuse_b=*/false);
    float s = c[0] + c[1] + c[2] + c[3] + c[4] + c[5] + c[6] + c[7];
    float tot = s + __shfl_xor(s, 16, 32);
    if (lane == 0) out[0] = -tot;
}

// ---------------------------------------------------------------------------
extern "C" void kernel_launch(void* const* d_in, const int* in_sizes, int n_in,
                              void* d_out, int out_size, void* d_ws, size_t ws_size,
                              hipStream_t stream)
{
    const float* y    = (const float*)d_in[0];
    const float* B1s1 = (const float*)d_in[1];
    const float* B1s2 = (const float*)d_in[2];
    const float* lam1 = (const float*)d_in[3];
    const float* lam2 = (const float*)d_in[4];
    const float* logq = (const float*)d_in[5];
    const float* logr = (const float*)d_in[6];
    const float* gi   = (const float*)d_in[7];
    const float* gc   = (const float*)d_in[8];

    float* partials = (float*)d_ws;      // 32 floats used

    const int block = 256;
    const int grid  = NSER / block;      // 32 blocks

    rskf_forward_kernel<<<grid, block, 0, stream>>>(
        y, B1s1, B1s2, lam1, lam2, logq, logr, gi, gc, partials);
    rskf_final_reduce<<<1, 32, 0, stream>>>(partials, (float*)d_out, grid);
}